// HiResPrecipNet_63814624084702
// MI455X (gfx1250) — compile-verified
//
#include <hip/hip_runtime.h>
#include <hip/hip_bf16.h>
#include <math.h>

// ---------------------------------------------------------------------------
// Types for WMMA
// ---------------------------------------------------------------------------
typedef __attribute__((ext_vector_type(2))) float v2f;
typedef __attribute__((ext_vector_type(8))) float v8f;

#define WMMA_F32_4(a, b, c) \
  __builtin_amdgcn_wmma_f32_16x16x4_f32(false, (a), false, (b), (short)0, (c), false, false)

// ---------------------------------------------------------------------------
// Monotonic float <-> u32 encoding for atomicMax-based segment_max
// ---------------------------------------------------------------------------
__device__ __forceinline__ unsigned int enc_f32(float f) {
  unsigned int b = __float_as_uint(f);
  return (b & 0x80000000u) ? ~b : (b | 0x80000000u);
}
__device__ __forceinline__ float dec_f32(unsigned int b) {
  b = (b & 0x80000000u) ? (b & 0x7FFFFFFFu) : ~b;
  return __uint_as_float(b);
}
#define ENC_NEG_INF 0x007FFFFFu  // enc_f32(-inf)

// ---------------------------------------------------------------------------
// Generic fill
// ---------------------------------------------------------------------------
__global__ void fill_u32_kernel(unsigned int* __restrict__ p, unsigned int v, long long n) {
  long long i = (long long)blockIdx.x * blockDim.x + threadIdx.x;
  if (i < n) p[i] = v;
}

// ---------------------------------------------------------------------------
// BatchNorm statistics, coalesced: blocks cover row-chunks, threads stripe
// columns (consecutive lanes -> consecutive addresses). Local register
// accumulation, then one atomicAdd per (block, column).
// ---------------------------------------------------------------------------
#define MAXCPT 3  // ceil(625 / 256)
__global__ void col_sum_kernel(const float* __restrict__ X, int N, int C, int rowsPerBlock,
                               float* __restrict__ sum, float* __restrict__ sumsq) {
  const int r0 = blockIdx.x * rowsPerBlock;
  const int r1 = (r0 + rowsPerBlock < N) ? (r0 + rowsPerBlock) : N;
  float s[MAXCPT] = {0.f, 0.f, 0.f};
  float q[MAXCPT] = {0.f, 0.f, 0.f};
  for (int r = r0; r < r1; ++r) {
    const float* __restrict__ row = X + (size_t)r * C;
    int ci = 0;
    for (int c = threadIdx.x; c < C; c += blockDim.x, ++ci) {
      float v = row[c];
      s[ci] += v;
      q[ci] += v * v;
    }
  }
  int ci = 0;
  for (int c = threadIdx.x; c < C; c += blockDim.x, ++ci) {
    atomicAdd(&sum[c], s[ci]);
    atomicAdd(&sumsq[c], q[ci]);
  }
}

__global__ void col_finalize_kernel(const float* __restrict__ sum, const float* __restrict__ sumsq,
                                    int N, int C, float* __restrict__ mean, float* __restrict__ rstd) {
  int c = blockIdx.x * blockDim.x + threadIdx.x;
  if (c >= C) return;
  float mu  = sum[c] / (float)N;
  float var = sumsq[c] / (float)N - mu * mu;
  mean[c] = mu;
  rstd[c] = rsqrtf(var + 1e-5f);
}

__global__ void bn_apply_kernel(const float* __restrict__ X, float* __restrict__ Y,
                                const float* __restrict__ mean, const float* __restrict__ rstd,
                                const float* __restrict__ g, const float* __restrict__ b,
                                long long total, int C, int do_relu) {
  long long i = (long long)blockIdx.x * blockDim.x + threadIdx.x;
  if (i >= total) return;
  int c = (int)(i % C);
  float y = (X[i] - mean[c]) * rstd[c] * g[c] + b[c];
  if (do_relu) y = fmaxf(y, 0.f);
  Y[i] = y;
}

// ---------------------------------------------------------------------------
// FAST GEMM: OUT[M,N] = X[M,K] @ W[K,N] (+bias) via V_WMMA_F32_16X16X4_F32.
// Requires: M % 16 == 0, N % 64 == 0. K arbitrary (tail handled once).
// One wave computes a 16x64 tile; incremented pointers keep the inner loop
// free of multiplies and branches.
// ---------------------------------------------------------------------------
__global__ void gemm_wmma_f32_fast(const float* __restrict__ X, const float* __restrict__ W,
                                   const float* __restrict__ bias, float* __restrict__ OUT,
                                   int M, int K, int N) {
  const int lane = threadIdx.x & 31;
  const int wave = threadIdx.x >> 5;
  const int tilesN = N >> 6;  // 64-wide column groups
  const int tilesM = M >> 4;
  const int tile = blockIdx.x * (blockDim.x >> 5) + wave;
  if (tile >= tilesM * tilesN) return;  // wave-uniform: EXEC all-1s for WMMA
  const int tm = tile / tilesN;
  const int tn = tile - tm * tilesN;
  const int row0 = tm << 4;
  const int col0 = tn << 6;
  const int half = lane >> 4;  // 0 or 1
  const int l16  = lane & 15;

  const float* __restrict__ ap = X + (size_t)(row0 + l16) * K + 2 * half;
  const float* __restrict__ bp = W + (size_t)(2 * half) * N + col0 + l16;
  const size_t bstep = (size_t)4 * N;

  v8f acc0 = {}, acc1 = {}, acc2 = {}, acc3 = {};
  const int Kmain = K & ~3;

  for (int k0 = 0; k0 < Kmain; k0 += 4) {
    v2f a;
    a.x = ap[0];
    a.y = ap[1];
    v2f b0, b1, b2, b3;
    b0.x = bp[0];  b0.y = bp[(size_t)N + 0];
    b1.x = bp[16]; b1.y = bp[(size_t)N + 16];
    b2.x = bp[32]; b2.y = bp[(size_t)N + 32];
    b3.x = bp[48]; b3.y = bp[(size_t)N + 48];
    acc0 = WMMA_F32_4(a, b0, acc0);
    acc1 = WMMA_F32_4(a, b1, acc1);
    acc2 = WMMA_F32_4(a, b2, acc2);
    acc3 = WMMA_F32_4(a, b3, acc3);
    ap += 4;
    bp += bstep;
  }

  if (Kmain < K) {  // K-tail (1..3 leftover), guarded loads, executed once
    const int ka = Kmain + 2 * half;
    const bool k0ok = (ka < K), k1ok = (ka + 1 < K);
    v2f a;
    a.x = k0ok ? ap[0] : 0.f;
    a.y = k1ok ? ap[1] : 0.f;
    v2f b0, b1, b2, b3;
    b0.x = k0ok ? bp[0]  : 0.f; b0.y = k1ok ? bp[(size_t)N + 0]  : 0.f;
    b1.x = k0ok ? bp[16] : 0.f; b1.y = k1ok ? bp[(size_t)N + 16] : 0.f;
    b2.x = k0ok ? bp[32] : 0.f; b2.y = k1ok ? bp[(size_t)N + 32] : 0.f;
    b3.x = k0ok ? bp[48] : 0.f; b3.y = k1ok ? bp[(size_t)N + 48] : 0.f;
    acc0 = WMMA_F32_4(a, b0, acc0);
    acc1 = WMMA_F32_4(a, b1, acc1);
    acc2 = WMMA_F32_4(a, b2, acc2);
    acc3 = WMMA_F32_4(a, b3, acc3);
  }

  const float bv0 = bias ? bias[col0 + l16 +  0] : 0.f;
  const float bv1 = bias ? bias[col0 + l16 + 16] : 0.f;
  const float bv2 = bias ? bias[col0 + l16 + 32] : 0.f;
  const float bv3 = bias ? bias[col0 + l16 + 48] : 0.f;
#pragma unroll
  for (int i = 0; i < 8; ++i) {
    const size_t r = (size_t)(row0 + i + 8 * half) * N + col0 + l16;
    OUT[r +  0] = acc0[i] + bv0;
    OUT[r + 16] = acc1[i] + bv1;
    OUT[r + 32] = acc2[i] + bv2;
    OUT[r + 48] = acc3[i] + bv3;
  }
}

// ---------------------------------------------------------------------------
// Guarded GEMM (any K, any N) — used only for the N=1 final layer.
// ---------------------------------------------------------------------------
__global__ void gemm_wmma_f32_guarded(const float* __restrict__ X, const float* __restrict__ W,
                                      const float* __restrict__ bias, float* __restrict__ OUT,
                                      int M, int K, int N) {
  const int lane = threadIdx.x & 31;
  const int wave = threadIdx.x >> 5;
  const int tilesN = (N + 15) >> 4;
  const int tilesM = M >> 4;
  const int tile = blockIdx.x * (blockDim.x >> 5) + wave;
  if (tile >= tilesM * tilesN) return;
  const int tm = tile / tilesN;
  const int tn = tile - tm * tilesN;
  const int row0 = tm << 4;
  const int col0 = tn << 4;
  const int half = lane >> 4;
  const int l16  = lane & 15;
  const int arow = row0 + l16;
  const int bcol = col0 + l16;
  const bool cok = (bcol < N);

  v8f acc = {};
  for (int k0 = 0; k0 < K; k0 += 4) {
    const int ka = k0 + 2 * half;
    v2f a, b;
    a.x = (ka     < K) ? X[(size_t)arow * K + ka    ] : 0.f;
    a.y = (ka + 1 < K) ? X[(size_t)arow * K + ka + 1] : 0.f;
    b.x = (cok && ka     < K) ? W[(size_t)(ka    ) * N + bcol] : 0.f;
    b.y = (cok && ka + 1 < K) ? W[(size_t)(ka + 1) * N + bcol] : 0.f;
    acc = WMMA_F32_4(a, b, acc);
  }

  if (cok) {
    float bv = bias ? bias[bcol] : 0.f;
#pragma unroll
    for (int i = 0; i < 8; ++i) {
      int r = row0 + i + 8 * half;
      OUT[(size_t)r * N + bcol] = acc[i] + bv;
    }
  }
}

// ---------------------------------------------------------------------------
// GATv2 edge kernels
// ---------------------------------------------------------------------------
// Wave-per-(edge,head): lanes stripe channels (coalesced), butterfly-reduce
// the leaky_relu(m)·att dot product, lane 0 writes score + segment-max.
__global__ void edge_scores_kernel(const float* __restrict__ xl, const float* __restrict__ xr,
                                   const int* __restrict__ src, const int* __restrict__ dst,
                                   const float* __restrict__ att,
                                   float* __restrict__ ebuf, unsigned int* __restrict__ emax_enc,
                                   int E, int H, int C) {
  const long long wid = ((long long)blockIdx.x * blockDim.x + threadIdx.x) >> 5;
  const int lane = threadIdx.x & 31;
  if (wid >= (long long)E * H) return;
  const int e = (int)(wid / H);
  const int h = (int)(wid % H);
  const int s = src[e];
  const int d = dst[e];
  const float* __restrict__ pl = xl + (size_t)s * H * C + (size_t)h * C;
  const float* __restrict__ pr = xr ? (xr + (size_t)d * H * C + (size_t)h * C) : nullptr;
  const float* __restrict__ pa = att + (size_t)h * C;
  float acc = 0.f;
  for (int c = lane; c < C; c += 32) {
    float m = pl[c] + (pr ? pr[c] : 0.f);
    float lr = (m > 0.f) ? m : 0.2f * m;
    acc += lr * pa[c];
  }
#pragma unroll
  for (int off = 16; off > 0; off >>= 1) acc += __shfl_xor(acc, off, 32);
  if (lane == 0) {
    ebuf[wid] = acc;
    atomicMax(&emax_enc[(size_t)d * H + h], enc_f32(acc));
  }
}

// ex = exp(e - emax[dst]); den[dst,h] += ex; cnt[dst] += 1 (once per edge)
__global__ void edge_exp_kernel(const float* __restrict__ ebuf, const int* __restrict__ dst,
                                const unsigned int* __restrict__ emax_enc,
                                float* __restrict__ den, float* __restrict__ cnt,
                                float* __restrict__ exbuf, int E, int H) {
  long long idx = (long long)blockIdx.x * blockDim.x + threadIdx.x;
  if (idx >= (long long)E * H) return;
  int e = (int)(idx / H);
  int h = (int)(idx % H);
  int d = dst[e];
  float mx = dec_f32(emax_enc[(size_t)d * H + h]);
  if (!(mx > -1e37f && mx < 1e37f)) mx = 0.f;  // match jnp.where(isfinite(emax), emax, 0)
  float ev = expf(ebuf[idx] - mx);
  exbuf[idx] = ev;
  atomicAdd(&den[(size_t)d * H + h], ev);
  if (h == 0) atomicAdd(&cnt[d], 1.f);
}

// out[dst,h,c] += xl[src,h,c] * ex[e,h] / (den[dst,h] + 1e-16)
__global__ void edge_aggregate_kernel(const float* __restrict__ xl,
                                      const int* __restrict__ src, const int* __restrict__ dst,
                                      const float* __restrict__ exbuf, const float* __restrict__ den,
                                      float* __restrict__ out, long long total, int H, int C) {
  long long idx = (long long)blockIdx.x * blockDim.x + threadIdx.x;
  if (idx >= total) return;
  const int hc = H * C;
  long long e = idx / hc;
  int r = (int)(idx % hc);
  int h = r / C;
  int s = src[e];
  int d = dst[e];
  float alpha = exbuf[e * H + h] / (den[(size_t)d * H + h] + 1e-16f);
  atomicAdd(&out[(size_t)d * hc + r], xl[(size_t)s * hc + r] * alpha);
}

// out /= max(cnt,1); out += bias
__global__ void finalize_kernel(float* __restrict__ out, const float* __restrict__ cnt,
                                const float* __restrict__ bias, long long total, int HC) {
  long long idx = (long long)blockIdx.x * blockDim.x + threadIdx.x;
  if (idx >= total) return;
  long long n = idx / HC;
  int j = (int)(idx % HC);
  float v = out[idx] / fmaxf(cnt[n], 1.f);
  if (bias) v += bias[j];
  out[idx] = v;
}

// encod_high = concat([z_std, enc128], axis=-1) -> [N,129]
__global__ void concat129_kernel(const float* __restrict__ z, const float* __restrict__ enc,
                                 float* __restrict__ out, int N) {
  long long idx = (long long)blockIdx.x * blockDim.x + threadIdx.x;
  if (idx >= (long long)N * 129) return;
  int n = (int)(idx / 129);
  int j = (int)(idx % 129);
  out[idx] = (j == 0) ? z[n] : enc[(size_t)n * 128 + (j - 1)];
}

__global__ void copy_f32_kernel(const float* __restrict__ s, float* __restrict__ d, int n) {
  int i = blockIdx.x * blockDim.x + threadIdx.x;
  if (i < n) d[i] = s[i];
}

// ---------------------------------------------------------------------------
// Host orchestration
// ---------------------------------------------------------------------------
extern "C" void kernel_launch(void* const* d_in, const int* in_sizes, int n_in,
                              void* d_out, int out_size, void* d_ws, size_t ws_size,
                              hipStream_t stream) {
  (void)in_sizes; (void)n_in; (void)out_size; (void)ws_size;

  const int NL = 10000, NH = 50000;
  const int EL = 160000, E2 = 400000, EH = 800000;

  const float* x_low  = (const float*)d_in[0];
  const float* z_std  = (const float*)d_in[1];
  const int*   ei_low = (const int*)d_in[2];
  const int*   ei_l2h = (const int*)d_in[3];
  const int*   ei_high= (const int*)d_in[4];
  auto P = [&](int i) { return (const float*)d_in[i]; };
  // param indices (setup_inputs insertion order):
  // 5:bn0.g 6:bn0.b 7:g1.Wl 8:g1.att 9:g1.Wr 10:g1.b 11:bn1.g 12:bn1.b
  // 13:g2.Wl 14:g2.att 15:g2.Wr 16:g2.b 17:bn2.g 18:bn2.b
  // 19:g3.Wl 20:g3.att 21:g3.Wr 22:g3.b 23:l2h.Wl 24:l2h.att
  // 25:bn3.g 26:bn3.b 27:g4.Wl 28:g4.att 29:g4.Wr 30:g4.b 31:bn4.g 32:bn4.b
  // 33:g5.Wl 34:g5.att 35:g5.Wr 36:g5.b 37:bn5.g 38:bn5.b
  // 39:g6.Wl 40:g6.att 41:g6.Wr 42:g6.b

  // ---- workspace bump allocator ----
  char* wsp = (char*)d_ws;
  auto alloc = [&](size_t bytes) -> char* {
    char* p = wsp;
    wsp += (bytes + 255) & ~(size_t)255;
    return p;
  };
  const size_t BIG = (size_t)NH * 256;  // 12.8M floats, max node-feature buffer
  float* B0 = (float*)alloc(BIG * sizeof(float));          // current features
  float* B1 = (float*)alloc(BIG * sizeof(float));          // xl
  float* B2 = (float*)alloc(BIG * sizeof(float));          // xr / l2h out
  float* B3 = (float*)alloc(BIG * sizeof(float));          // gat output accumulator
  float* ebuf  = (float*)alloc((size_t)EH * 2 * sizeof(float));
  float* exbuf = (float*)alloc((size_t)EH * 2 * sizeof(float));
  unsigned int* emax = (unsigned int*)alloc((size_t)NH * 2 * sizeof(unsigned int));
  float* den  = (float*)alloc((size_t)NH * 2 * sizeof(float));
  float* cnt  = (float*)alloc((size_t)NH * sizeof(float));
  float* csum   = (float*)alloc(1024 * sizeof(float));
  float* csumsq = (float*)alloc(1024 * sizeof(float));
  float* mean = (float*)alloc(1024 * sizeof(float));
  float* rstd = (float*)alloc(1024 * sizeof(float));

  auto fill = [&](void* p, unsigned int v, long long n) {
    fill_u32_kernel<<<(unsigned)((n + 255) / 256), 256, 0, stream>>>((unsigned int*)p, v, n);
  };
  auto bn = [&](const float* x, float* y, const float* g, const float* b,
                int N, int C, int relu) {
    fill(csum, 0u, C);
    fill(csumsq, 0u, C);
    const int rpb = 128;
    const int nb = (N + rpb - 1) / rpb;
    col_sum_kernel<<<nb, 256, 0, stream>>>(x, N, C, rpb, csum, csumsq);
    col_finalize_kernel<<<(C + 255) / 256, 256, 0, stream>>>(csum, csumsq, N, C, mean, rstd);
    long long total = (long long)N * C;
    bn_apply_kernel<<<(unsigned)((total + 255) / 256), 256, 0, stream>>>(
        x, y, mean, rstd, g, b, total, C, relu);
  };
  auto gemm = [&](const float* X, const float* W, const float* bias, float* O,
                  int M, int K, int N) {
    if ((N & 63) == 0) {
      int tiles = (M >> 4) * (N >> 6);
      gemm_wmma_f32_fast<<<(unsigned)((tiles + 7) / 8), 256, 0, stream>>>(X, W, bias, O, M, K, N);
    } else {
      int tiles = (M >> 4) * ((N + 15) >> 4);
      gemm_wmma_f32_guarded<<<(unsigned)((tiles + 7) / 8), 256, 0, stream>>>(X, W, bias, O, M, K, N);
    }
  };
  auto gat = [&](const float* xsrc, const float* xdst, const int* ei, int E,
                 int Nsrc, int Ndst, int Cin, int H, int C,
                 const float* Wl, const float* att, const float* Wr, const float* bias,
                 float* xl, float* xr, float* out) {
    gemm(xsrc, Wl, nullptr, xl, Nsrc, Cin, H * C);
    if (Wr) gemm(xdst, Wr, nullptr, xr, Ndst, Cin, H * C);
    fill(emax, ENC_NEG_INF, (long long)Ndst * H);
    fill(den, 0u, (long long)Ndst * H);
    fill(cnt, 0u, Ndst);
    fill(out, 0u, (long long)Ndst * H * C);
    const int* src = ei;
    const int* dst = ei + E;
    long long waves = (long long)E * H;  // one wave per (edge, head)
    edge_scores_kernel<<<(unsigned)((waves + 7) / 8), 256, 0, stream>>>(
        xl, Wr ? xr : nullptr, src, dst, att, ebuf, emax, E, H, C);
    long long eh = (long long)E * H;
    edge_exp_kernel<<<(unsigned)((eh + 255) / 256), 256, 0, stream>>>(
        ebuf, dst, emax, den, cnt, exbuf, E, H);
    long long tot = (long long)E * H * C;
    edge_aggregate_kernel<<<(unsigned)((tot + 255) / 256), 256, 0, stream>>>(
        xl, src, dst, exbuf, den, out, tot, H, C);
    long long nhc = (long long)Ndst * H * C;
    finalize_kernel<<<(unsigned)((nhc + 255) / 256), 256, 0, stream>>>(
        out, cnt, bias, nhc, H * C);
  };

  // ---------------- low net ----------------
  bn(x_low, B0, P(5), P(6), NL, 625, 0);                                   // bn0
  gat(B0, B0, ei_low, EL, NL, NL, 625, 1, 512,
      P(7), P(8), P(9), P(10), B1, B2, B3);                                 // gat1 -> B3
  bn(B3, B0, P(11), P(12), NL, 512, 1);                                     // bn1 + relu
  gat(B0, B0, ei_low, EL, NL, NL, 512, 1, 512,
      P(13), P(14), P(15), P(16), B1, B2, B3);                              // gat2 -> B3
  bn(B3, B0, P(17), P(18), NL, 512, 1);                                     // bn2 + relu
  gat(B0, B0, ei_low, EL, NL, NL, 512, 1, 512,
      P(19), P(20), P(21), P(22), B1, B2, B3);                              // gat3 (encod_low) -> B3

  // ---------------- low -> high bipartite (no Wr, no bias) ----------------
  gat(B3, nullptr, ei_l2h, E2, NL, NH, 512, 1, 128,
      P(23), P(24), nullptr, nullptr, B1, nullptr, B2);                     // l2h -> B2 [NH,128]

  // concat z_std with l2h output -> B0 [NH,129]
  {
    long long tot = (long long)NH * 129;
    concat129_kernel<<<(unsigned)((tot + 255) / 256), 256, 0, stream>>>(z_std, B2, B0, NH);
  }

  // ---------------- high net ----------------
  bn(B0, B0, P(25), P(26), NH, 129, 0);                                     // bn3 (in place)
  gat(B0, B0, ei_high, EH, NH, NH, 129, 2, 128,
      P(27), P(28), P(29), P(30), B1, B2, B3);                              // gat4 (2 heads) -> B3 [NH,256]
  bn(B3, B0, P(31), P(32), NH, 256, 1);                                     // bn4 + relu
  gat(B0, B0, ei_high, EH, NH, NH, 256, 1, 128,
      P(33), P(34), P(35), P(36), B1, B2, B3);                              // gat5 -> B3 [NH,128]
  bn(B3, B0, P(37), P(38), NH, 128, 1);                                     // bn5 + relu
  gat(B0, B0, ei_high, EH, NH, NH, 128, 1, 1,
      P(39), P(40), P(41), P(42), B1, B2, B3);                              // gat6 -> B3 [NH,1]

  copy_f32_kernel<<<(NH + 255) / 256, 256, 0, stream>>>(B3, (float*)d_out, NH);
}